// ScaledDotProductAttention_7421703487710
// MI455X (gfx1250) — compile-verified
//
#include <hip/hip_runtime.h>
#include <stdint.h>

typedef float v2f __attribute__((ext_vector_type(2)));
typedef float v4f __attribute__((ext_vector_type(4)));
typedef float v8f __attribute__((ext_vector_type(8)));
typedef int   v4i __attribute__((ext_vector_type(4)));

#define BH 64
#define L 1024
#define D 64
#define MB 32            // rows per workgroup
#define NTHREADS 256     // 8 waves (wave32)
#define INV_T 0.125f
#define NEGVAL -1e15f
#define SQ_STRIDE 68     // mod 64 = 4 -> conflict-free A-fragment LDS reads
#define SS_STRIDE 1028   // mod 64 = 4 -> conflict-free P-fragment LDS reads

__global__ __launch_bounds__(NTHREADS) void attn_fp32_wmma_kernel(
    const float* __restrict__ q, const float* __restrict__ k,
    const float* __restrict__ v, const uint8_t* __restrict__ mask,
    const int* __restrict__ edge, float* __restrict__ out_o,
    float* __restrict__ out_attn)
{
    __shared__ float sQ[MB * SQ_STRIDE];   // ~8.7 KB
    __shared__ float sS[MB * SS_STRIDE];   // ~131.6 KB  (total ~140 KB, 2 WGs/WGP)

    const int tid    = threadIdx.x;
    const int lane   = tid & 31;
    const int wave   = tid >> 5;           // 0..7
    const int lane16 = lane & 15;          // M (A/C) or N (B) index
    const int koff   = (lane >> 4) << 1;   // lanes 16-31 hold K+2,K+3
    const int m0     = blockIdx.x * MB;    // 32-row block
    const int bh     = blockIdx.y;

    const float* qb = q + ((size_t)bh * L + m0) * D;
    const float* kb = k + (size_t)bh * L * D;
    const float* vb = v + (size_t)bh * L * D;

    // ---- stage Q tile (32x64) into LDS, coalesced b128 ----
    for (int idx = tid; idx < MB * D / 4; idx += NTHREADS) {
        int r = idx >> 4, c4 = (idx & 15) << 2;      // 16 vec4 per row
        *(v4f*)&sQ[r * SQ_STRIDE + c4] = *(const v4f*)&qb[r * D + c4];
    }
    __syncthreads();

    // ---- Phase A: S = Q * K^T via V_WMMA_F32_16X16X4_F32 ----
    // 2 row-tiles x 64 col-tiles = 128 jobs over 8 waves
    for (int job = wave; job < 2 * (L / 16); job += 8) {
        const int rt = job >> 6;           // row tile 0/1
        const int n0 = (job & 63) << 4;    // col tile base
        v8f acc = {};
        const float* qrow = sQ + (rt * 16 + lane16) * SQ_STRIDE + koff;  // A: Q[m][k]
        const float* krow = kb + (size_t)(n0 + lane16) * D + koff;       // B: K[n][k]
#pragma unroll
        for (int k0 = 0; k0 < D; k0 += 4) {
            v2f a; a.x = qrow[k0]; a.y = qrow[k0 + 1];
            v2f b; b.x = krow[k0]; b.y = krow[k0 + 1];
            acc = __builtin_amdgcn_wmma_f32_16x16x4_f32(
                false, a, false, b, (short)0, acc, false, false);
        }
        const int rbase = rt * 16 + ((lane >> 4) << 3); // lanes16-31 hold M=8..15
#pragma unroll
        for (int vv = 0; vv < 8; ++vv)
            sS[(rbase + vv) * SS_STRIDE + n0 + lane16] = acc[vv];
    }
    __syncthreads();

    // ---- Phase B: scale + edge/mask; vectorized, non-temporal (read-once) ----
    {
        const int b4 = bh & 3;             // jnp.tile(edge,(16,1,1)) -> edge[bh % 4]
        const uint8_t* mrow = mask + ((size_t)bh * L + m0) * L;
        const int*     erow = edge + ((size_t)b4 * L + m0) * L;
        for (int idx = tid; idx < MB * L / 4; idx += NTHREADS) {
            int r = idx >> 8, c4 = (idx & 255) << 2;   // 256 vec4 per row
            size_t off = (size_t)r * L + c4;
            v4f val = *(v4f*)&sS[r * SS_STRIDE + c4];
            v4i e4 = __builtin_nontemporal_load((const v4i*)(erow + off));
            uint32_t m4 = __builtin_nontemporal_load((const uint32_t*)(mrow + off));
#pragma unroll
            for (int j = 0; j < 4; ++j) {
                float s = val[j] * INV_T;
                if (e4[j] == 0 || ((m4 >> (8 * j)) & 0xFF)) s = NEGVAL;
                val[j] = s;
            }
            *(v4f*)&sS[r * SS_STRIDE + c4] = val;
        }
    }
    __syncthreads();

    // ---- Phase C: row softmax, 8-lane team per row, vectorized x4 ----
    {
        const int row = tid >> 3;          // 0..31
        const int l8  = tid & 7;
        float* srow = sS + row * SS_STRIDE;
        float mx = -3.4e38f;
        for (int i = 0; i < L / 32; ++i) {
            v4f s = *(const v4f*)&srow[(i * 8 + l8) * 4];
            mx = fmaxf(mx, fmaxf(fmaxf(s[0], s[1]), fmaxf(s[2], s[3])));
        }
        mx = fmaxf(mx, __shfl_xor(mx, 4, 8));
        mx = fmaxf(mx, __shfl_xor(mx, 2, 8));
        mx = fmaxf(mx, __shfl_xor(mx, 1, 8));
        float sum = 0.f;
        for (int i = 0; i < L / 32; ++i) {
            v4f s = *(v4f*)&srow[(i * 8 + l8) * 4];
#pragma unroll
            for (int j = 0; j < 4; ++j) { s[j] = __expf(s[j] - mx); sum += s[j]; }
            *(v4f*)&srow[(i * 8 + l8) * 4] = s;
        }
        sum += __shfl_xor(sum, 4, 8);
        sum += __shfl_xor(sum, 2, 8);
        sum += __shfl_xor(sum, 1, 8);
        const float rinv = 1.0f / sum;
        for (int i = 0; i < L / 32; ++i) {
            v4f s = *(v4f*)&srow[(i * 8 + l8) * 4];
            s *= rinv;
            *(v4f*)&srow[(i * 8 + l8) * 4] = s;
        }
    }
    __syncthreads();

    // ---- Phase D: stream attn out; b128 non-temporal stores (don't thrash L2) ----
    {
        float* arow = out_attn + ((size_t)bh * L + m0) * L;
        for (int idx = tid; idx < MB * L / 4; idx += NTHREADS) {
            int r = idx >> 8, c4 = (idx & 255) << 2;
            v4f val = *(const v4f*)&sS[r * SS_STRIDE + c4];
            __builtin_nontemporal_store(val, (v4f*)(arow + (size_t)r * L + c4));
        }
    }

    // ---- Phase E: O = P * V via WMMA; 2 row-tiles x 4 D-tiles = 1 job/wave ----
    {
        const int rt = wave >> 2;
        const int d0 = (wave & 3) << 4;
        v8f acc = {};
        const float* prow = sS + (rt * 16 + lane16) * SS_STRIDE + koff;  // A: P[m][k]
        const float* vcol = vb + d0 + lane16;                            // B: V[k][n]
        for (int k0 = 0; k0 < L; k0 += 4) {
            v2f a; a.x = prow[k0]; a.y = prow[k0 + 1];
            v2f b;
            b.x = vcol[(size_t)(k0 + koff) * D];
            b.y = vcol[(size_t)(k0 + koff + 1) * D];
            acc = __builtin_amdgcn_wmma_f32_16x16x4_f32(
                false, a, false, b, (short)0, acc, false, false);
        }
        const int rbase = (lane >> 4) << 3;
        float* orow = out_o + ((size_t)bh * L + m0 + rt * 16) * D + d0 + lane16;
#pragma unroll
        for (int vv = 0; vv < 8; ++vv)
            orow[(size_t)(rbase + vv) * D] = acc[vv];
    }
}

extern "C" void kernel_launch(void* const* d_in, const int* in_sizes, int n_in,
                              void* d_out, int out_size, void* d_ws, size_t ws_size,
                              hipStream_t stream) {
    const float*   q    = (const float*)d_in[0];
    const float*   k    = (const float*)d_in[1];
    const float*   v    = (const float*)d_in[2];
    const uint8_t* mask = (const uint8_t*)d_in[3]; // jnp.bool_ -> 1 byte
    const int*     edge = (const int*)d_in[4];

    float* out_o    = (float*)d_out;
    float* out_attn = out_o + (size_t)BH * L * D;

    dim3 grid(L / MB, BH);
    attn_fp32_wmma_kernel<<<grid, NTHREADS, 0, stream>>>(q, k, v, mask, edge,
                                                         out_o, out_attn);
}